// KANCubicLayer_8727373546283
// MI455X (gfx1250) — compile-verified
//
#include <hip/hip_runtime.h>

// ---------------------------------------------------------------------------
// KAN cubic-spline layer on MI455X (gfx1250), recast as two f32 WMMA GEMMs:
//   Y^T[o,b] = sum_q coeffs[o,q]*Wc[q,b] + sum_q slopes[o,q]*Wv[q,b] + bias[o]
// q = (j,k) in [0, 8192). coeffs is ALREADY (o,q)-major -> A operand needs no
// transpose. slopes computed by PCHIP prolog in the same layout. Wc/Wv are the
// Hermite-basis / extrapolation weights (sparse in k), written dense.
// Roofline: 4.3 GFLOP GEMM; ~50MB working set fits L2 (192MB); ~64MB HBM
// traffic ~ 2.7us @ 23.3 TB/s -> matrix-pipe bound -> V_WMMA_F32_16X16X4_F32
// (f32 keeps full precision of O(1e-3) spline coefficients).
// CDNA5 paths: WMMA f32, GLOBAL_LOAD_ASYNC_TO_LDS_B128 (ASYNCcnt) for the
// B-operand LDS staging, GLOBAL_PREFETCH for the A stream. Inner K-tile is
// software-pipelined: all A/B fragments preloaded so the loads clause up with
// staggered counter waits instead of a wait-0 stall before every WMMA.
// ---------------------------------------------------------------------------

#define KAN_B    256
#define KAN_DIN  512
#define KAN_DOUT 512
#define KAN_K    16
#define KAN_QD   (KAN_DIN * KAN_K)   // 8192 per table
#define KAN_EPS  1e-12f
#define OUT_N    (KAN_B * KAN_DOUT)  // 131072

typedef float v2f __attribute__((ext_vector_type(2)));
typedef float v4f __attribute__((ext_vector_type(4)));
typedef float v8f __attribute__((ext_vector_type(8)));
typedef int   v4i __attribute__((ext_vector_type(4)));

#if defined(__gfx1250__) && \
    __has_builtin(__builtin_amdgcn_global_load_async_to_lds_b128) && \
    __has_builtin(__builtin_amdgcn_s_wait_asynccnt)
#define USE_ASYNC_LDS 1
#define AS1 __attribute__((address_space(1)))
#define AS3 __attribute__((address_space(3)))

// Builtin signature (from hipcc diagnostic): (int4 AS1*, int4 AS3*, imm, imm).
__device__ inline AS1 v4i* kan_to_global_v4i(const void* p) {
  return (AS1 v4i*)(v4i*)(uintptr_t)p;   // strip const, then generic->AS1
}
__device__ inline AS3 v4i* kan_to_shared_v4i(void* p) {
  return (AS3 v4i*)(v4i*)p;              // generic->AS3 (p points into LDS)
}
#endif

// ---------------------------------------------------------------------------
// Kernel 1: PCHIP slopes. One thread per (o,j) row of 16 knots.
// ---------------------------------------------------------------------------
__global__ void kan_slopes(const float* __restrict__ coeffs,
                           const float* __restrict__ knots,
                           float* __restrict__ slopes) {
  const int row = blockIdx.x * blockDim.x + threadIdx.x;
  if (row >= KAN_DOUT * KAN_DIN) return;
  const float* src = coeffs + (size_t)row * KAN_K;

  float y[16];
  *(v4f*)&y[0]  = *(const v4f*)(src + 0);
  *(v4f*)&y[4]  = *(const v4f*)(src + 4);
  *(v4f*)&y[8]  = *(const v4f*)(src + 8);
  *(v4f*)&y[12] = *(const v4f*)(src + 12);

  float kn[16];
#pragma unroll
  for (int i = 0; i < 16; ++i) kn[i] = knots[i];

  float h[15], del[15];
#pragma unroll
  for (int i = 0; i < 15; ++i) {
    h[i]   = kn[i + 1] - kn[i];
    del[i] = (y[i + 1] - y[i]) / (h[i] + KAN_EPS);
  }

  float d[16];
#pragma unroll
  for (int i = 1; i < 15; ++i) {
    const float h0 = h[i - 1], h1 = h[i];
    const float w1 = 2.f * h1 + h0, w2 = h1 + 2.f * h0;
    const float di = (w1 + w2) /
        (w1 / (del[i - 1] + KAN_EPS) + w2 / (del[i] + KAN_EPS) + KAN_EPS);
    d[i] = (del[i - 1] * del[i] > 0.f) ? di : 0.f;
  }
  // left endpoint
  float d0 = ((2.f * h[0] + h[1]) * del[0] - h[0] * del[1]) /
             (h[0] + h[1] + KAN_EPS);
  d0 = (d0 * del[0] <= 0.f) ? 0.f : d0;
  d0 = ((del[0] * del[1] < 0.f) && (fabsf(d0) > 3.f * fabsf(del[0])))
           ? 3.f * del[0] : d0;
  d[0] = d0;
  // right endpoint
  float dn = ((2.f * h[14] + h[13]) * del[14] - h[14] * del[13]) /
             (h[14] + h[13] + KAN_EPS);
  dn = (dn * del[14] <= 0.f) ? 0.f : dn;
  dn = ((del[14] * del[13] < 0.f) && (fabsf(dn) > 3.f * fabsf(del[14])))
           ? 3.f * del[14] : dn;
  d[15] = dn;

  float* dst = slopes + (size_t)row * KAN_K;
  *(v4f*)(dst + 0)  = *(v4f*)&d[0];
  *(v4f*)(dst + 4)  = *(v4f*)&d[4];
  *(v4f*)(dst + 8)  = *(v4f*)&d[8];
  *(v4f*)(dst + 12) = *(v4f*)&d[12];
}

// ---------------------------------------------------------------------------
// Kernel 2: Hermite/extrapolation weights in (q,b) layout.
// blockIdx.x = j, threadIdx.x = b -> all 32 stores coalesced across lanes.
// ---------------------------------------------------------------------------
__global__ void kan_weights(const float* __restrict__ x,
                            const float* __restrict__ knots,
                            float* __restrict__ Wc,
                            float* __restrict__ Wv) {
  const int b = threadIdx.x;   // 0..255
  const int j = blockIdx.x;    // 0..511
  const float xv = x[(size_t)b * KAN_DIN + j];
  const float x0 = knots[0], xN = knots[KAN_K - 1];
  const float dx = (xN - x0) / (float)(KAN_K - 1);
  const float u  = (xv - x0) / (dx + KAN_EPS);
  int idx = (int)floorf(u);
  idx = idx < 0 ? 0 : (idx > KAN_K - 2 ? KAN_K - 2 : idx);
  const float t  = u - (float)idx;
  const bool left  = xv < x0;
  const bool right = xv > xN;
  const float t2 = t * t, t3 = t2 * t;
  const float h00 = 2.f * t3 - 3.f * t2 + 1.f;
  const float h01 = -2.f * t3 + 3.f * t2;
  const float h10 = (t3 - 2.f * t2 + t) * dx;
  const float h11 = (t3 - t2) * dx;

  float* pc = Wc + (size_t)j * KAN_K * KAN_B + b;
  float* pv = Wv + (size_t)j * KAN_K * KAN_B + b;
#pragma unroll
  for (int k = 0; k < KAN_K; ++k) {
    const float wcm = (k == idx) ? h00 : (k == idx + 1) ? h01 : 0.f;
    const float wvm = (k == idx) ? h10 : (k == idx + 1) ? h11 : 0.f;
    const float wck = left  ? ((k == 0) ? 1.f : 0.f)
                    : right ? ((k == KAN_K - 1) ? 1.f : 0.f)
                            : wcm;
    const float wvk = left  ? ((k == 0) ? (xv - x0) : 0.f)
                    : right ? ((k == KAN_K - 1) ? (xv - xN) : 0.f)
                            : wvm;
    pc[(size_t)k * KAN_B] = wck;
    pv[(size_t)k * KAN_B] = wvk;
  }
}

// ---------------------------------------------------------------------------
// Kernel 3: WMMA GEMM. Block: 256 thr / 8 waves, tile 64(o) x 64(b); wave tile
// 16(o) x 32(b) = two f32 16x16 accumulators. K split x4 via blockIdx.z
// (splits 0,1 -> coeffs*Wc, 2,3 -> slopes*Wv; 4096 q each), partials to ws.
// W tile (64 q-rows x 64 b, row stride 80 floats -> half-wave rows r / r+2 hit
// disjoint bank groups) staged in LDS via GLOBAL_LOAD_ASYNC_TO_LDS_B128.
// A loaded per-lane as float2 per the 16x4 f32 A layout (lanes 0-15: K0/K1,
// lanes 16-31: K2/K3). All fragments for a tile are preloaded to registers
// (software pipeline) so vmem/ds latency overlaps the WMMA chain.
// ---------------------------------------------------------------------------
#define LDS_STRIDE 80
#define QI 64
#define NSTEP (QI / 4)   // 16 WMMA K-steps per LDS tile

__global__ void kan_gemm(const float* __restrict__ coeffs,
                         const float* __restrict__ slopes,
                         const float* __restrict__ Wc,
                         const float* __restrict__ Wv,
                         float* __restrict__ part) {
  __shared__ float ldsW[QI * LDS_STRIDE];   // 20 KB

  const int s = blockIdx.z;                       // 0..3
  const float* tab = (s < 2) ? coeffs : slopes;   // A operand (o,q)-major
  const float* Wt  = (s < 2) ? Wc : Wv;           // B operand (q,b)-major
  const int qsplit = (s & 1) * 4096;

  const int oBlock = blockIdx.x * 64;
  const int bBlock = blockIdx.y * 64;
  const int tid  = threadIdx.x;
  const int w    = tid >> 5;
  const int lane = tid & 31;
  const int ln   = lane & 15;
  const int kp   = (lane >> 4) << 1;              // 0 or 2 (K-pair select)
  const int oWave = oBlock + (w & 3) * 16;
  const int bWave = (w >> 2) * 32;                // offset inside block tile

  v8f acc0 = {};
  v8f acc1 = {};

  const float* aRow = tab + (size_t)(oWave + ln) * KAN_QD + qsplit + kp;
  const int col0 = bWave + ln;
  const int col1 = col0 + 16;

  for (int qc = 0; qc < 4096; qc += QI) {
    __syncthreads();  // previous tile fully consumed by all waves
#pragma unroll
    for (int rep = 0; rep < 4; ++rep) {
      const int idx = tid + rep * 256;            // 0..1023 float4 chunks
      const int r = idx >> 4;                     // 0..63 q-row
      const int c = (idx & 15) << 2;              // 0..60 b-col
      const float* gsrc = Wt + (size_t)(qsplit + qc + r) * KAN_B + bBlock + c;
      float* ldst = &ldsW[r * LDS_STRIDE + c];
#ifdef USE_ASYNC_LDS
      __builtin_amdgcn_global_load_async_to_lds_b128(
          kan_to_global_v4i(gsrc), kan_to_shared_v4i(ldst), 0, 0);
#else
      *(v4f*)ldst = *(const v4f*)gsrc;
#endif
    }
#ifdef USE_ASYNC_LDS
    __builtin_amdgcn_s_wait_asynccnt(0);
#endif
    __syncthreads();

    // Speculative prefetch of next A chunk (fault-free past the end).
    __builtin_prefetch(aRow + qc + QI, 0, 3);

    // --- Stage 1: preload all A fragments (global, L2-resident) -----------
    v2f af[NSTEP];
#pragma unroll
    for (int kk = 0; kk < NSTEP; ++kk)
      af[kk] = *(const v2f*)(aRow + qc + kk * 4);

    // --- Stage 2: preload all B fragments (LDS) ---------------------------
    v2f b0f[NSTEP], b1f[NSTEP];
#pragma unroll
    for (int kk = 0; kk < NSTEP; ++kk) {
      const int rb = kk * 4 + kp;
      v2f t0 = { ldsW[(rb + 0) * LDS_STRIDE + col0],
                 ldsW[(rb + 1) * LDS_STRIDE + col0] };
      v2f t1 = { ldsW[(rb + 0) * LDS_STRIDE + col1],
                 ldsW[(rb + 1) * LDS_STRIDE + col1] };
      b0f[kk] = t0;
      b1f[kk] = t1;
    }

    // --- Stage 3: WMMA chain ----------------------------------------------
#pragma unroll
    for (int kk = 0; kk < NSTEP; ++kk) {
      acc0 = __builtin_amdgcn_wmma_f32_16x16x4_f32(
          false, af[kk], false, b0f[kk], (short)0, acc0, false, false);
      acc1 = __builtin_amdgcn_wmma_f32_16x16x4_f32(
          false, af[kk], false, b1f[kk], (short)0, acc1, false, false);
    }
  }

  // D layout: VGPR r holds rows r (lanes 0-15) / r+8 (lanes 16-31), N = lane&15.
  // Store [s][b][o]: per lane two contiguous float4 runs along o.
  float* pS = part + (size_t)s * OUT_N;
  const int oBase = oWave + ((lane >> 4) << 3);
  const int bg0 = bBlock + bWave + ln;
  {
    float* p = pS + (size_t)bg0 * KAN_DOUT + oBase;
    v4f lo = { acc0[0], acc0[1], acc0[2], acc0[3] };
    v4f hi = { acc0[4], acc0[5], acc0[6], acc0[7] };
    *(v4f*)p = lo;
    *(v4f*)(p + 4) = hi;
  }
  {
    float* p = pS + (size_t)(bg0 + 16) * KAN_DOUT + oBase;
    v4f lo = { acc1[0], acc1[1], acc1[2], acc1[3] };
    v4f hi = { acc1[4], acc1[5], acc1[6], acc1[7] };
    *(v4f*)p = lo;
    *(v4f*)(p + 4) = hi;
  }
}

// ---------------------------------------------------------------------------
// Kernel 4: deterministic fixed-order reduction of the 4 K-split partials
// plus bias. part layout [s][b][o] matches out [b][o] -> fully coalesced.
// ---------------------------------------------------------------------------
__global__ void kan_reduce(const float* __restrict__ part,
                           const float* __restrict__ bias,
                           float* __restrict__ out) {
  const int i = blockIdx.x * blockDim.x + threadIdx.x;
  if (i >= OUT_N) return;
  const int o = i & (KAN_DOUT - 1);
  float v = bias[o];
  v += part[i];
  v += part[OUT_N + i];
  v += part[2 * OUT_N + i];
  v += part[3 * OUT_N + i];
  out[i] = v;
}

// ---------------------------------------------------------------------------
extern "C" void kernel_launch(void* const* d_in, const int* in_sizes, int n_in,
                              void* d_out, int out_size, void* d_ws, size_t ws_size,
                              hipStream_t stream) {
  const float* x      = (const float*)d_in[0];   // (256, 512)
  const float* knots  = (const float*)d_in[1];   // (16,)
  const float* coeffs = (const float*)d_in[2];   // (512, 512, 16)
  const float* bias   = (const float*)d_in[3];   // (512,)
  float* out = (float*)d_out;                    // (256, 512)

  // Workspace layout (34 MB total):
  float* slopes = (float*)d_ws;                                  // 16 MB
  float* Wc     = slopes + (size_t)KAN_DOUT * KAN_DIN * KAN_K;   //  8 MB
  float* Wv     = Wc + (size_t)KAN_QD * KAN_B;                   //  8 MB
  float* part   = Wv + (size_t)KAN_QD * KAN_B;                   //  2 MB

  // 1) PCHIP slopes: 262144 rows
  kan_slopes<<<(KAN_DOUT * KAN_DIN) / 256, 256, 0, stream>>>(coeffs, knots, slopes);

  // 2) Spline weights: grid over j, lanes over b (coalesced W writes)
  kan_weights<<<KAN_DIN, KAN_B, 0, stream>>>(x, knots, Wc, Wv);

  // 3) WMMA GEMM: (o-blocks, b-blocks, k-splits) = (8, 4, 4)
  dim3 ggrid(KAN_DOUT / 64, KAN_B / 64, 4);
  kan_gemm<<<ggrid, 256, 0, stream>>>(coeffs, slopes, Wc, Wv, part);

  // 4) Fixed-order reduction + bias
  kan_reduce<<<OUT_N / 256, 256, 0, stream>>>(part, bias, out);
}